// graphattention_80590766342925
// MI455X (gfx1250) — compile-verified
//
#include <hip/hip_runtime.h>

// ---------------- problem constants (from reference) ----------------
#define BB    8
#define NN    2048
#define FF    512
#define BN    (BB * NN)          // 16384 rows total
#define HEADS 8
#define NEG_SLOPE 0.01f
#define MASK_VAL  -1.0e10f
#define KBLK  32                 // softmax/GEMM K block in attn kernel

typedef __attribute__((ext_vector_type(2))) float v2f;
typedef __attribute__((ext_vector_type(8))) float v8f;
typedef __attribute__((ext_vector_type(4))) unsigned int u32x4;
typedef __attribute__((ext_vector_type(8))) int i32x8;
typedef __attribute__((ext_vector_type(4))) int i32x4;

#if defined(__has_builtin)
# if __has_builtin(__builtin_amdgcn_tensor_load_to_lds)
#  define HAVE_TDM 1
# endif
#endif
#ifndef HAVE_TDM
# define HAVE_TDM 0
#endif

static __device__ __forceinline__ v8f wmma4(v2f a, v2f b, v8f c) {
  // V_WMMA_F32_16X16X4_F32 : D = A(16x4,f32) x B(4x16,f32) + C(16x16,f32)
  return __builtin_amdgcn_wmma_f32_16x16x4_f32(
      false, a, false, b, (short)0, c, false, false);
}

#if HAVE_TDM
// Issue a TDM load of a [KBLK x FF] f32 tile (row-major, row stride FF elems)
// from global memory into LDS at byte offset lds_off. 2D descriptor per
// CDNA5 ISA (count=1, type=2 "image",
// data_size=4B, tile_dim0=FF, tile_dim1=KBLK, tensor_dim0_stride=FF).
static __device__ __forceinline__ void tdm_load_feat_tile(const float* gptr,
                                                          unsigned lds_off) {
  const unsigned long long ga = (unsigned long long)(uintptr_t)gptr;
  u32x4 g0;
  g0[0] = 1u;                                        // count=1 (valid), user mode
  g0[1] = lds_off;                                   // LDS byte address
  g0[2] = (unsigned)(ga & 0xFFFFFFFFu);              // global_addr[31:0]
  g0[3] = (unsigned)((ga >> 32) & 0x1FFFFFFu)        // global_addr[56:32]
          | (2u << 30);                              // type=2
  i32x8 g1;
  g1[0] = 0x00020000;                                // wg_mask=0, data_size=2 (4B)
  g1[1] = (int)(((unsigned)FF & 0xFFFFu) << 16);     // tensor_dim0 lo16 @ [63:48]
  g1[2] = (int)(((unsigned)NN & 0xFFFFu) << 16);     // tensor_dim1 lo16 @ [95:80]
  g1[3] = (int)(((unsigned)FF & 0xFFFFu) << 16);     // tile_dim0 @ [127:112]
  g1[4] = KBLK;                                      // tile_dim1=32, tile_dim2=0
  g1[5] = FF;                                        // tensor_dim0_stride lo32
  g1[6] = 0;
  g1[7] = 0;
  const i32x4 z4 = {0, 0, 0, 0};
#if __clang_major__ >= 23
  const i32x8 z8 = {0, 0, 0, 0, 0, 0, 0, 0};
  __builtin_amdgcn_tensor_load_to_lds(g0, g1, z4, z4, z8, 0);
#else
  __builtin_amdgcn_tensor_load_to_lds(g0, g1, z4, z4, 0);
#endif
}
#endif  // HAVE_TDM

// ---------------------------------------------------------------
// Kernel 1: feat = X @ Wk + bk     [BN x FF] = [BN x FF][FF x FF]
// block (32,8): 8 waves; each wave -> one 16x16 tile; grid (BN/16, FF/128)
// ---------------------------------------------------------------
__global__ __launch_bounds__(256) void proj_kernel(
    const float* __restrict__ X, const float* __restrict__ Wk,
    const float* __restrict__ bk, float* __restrict__ feat) {
  const int tx = threadIdx.x;            // 0..31 (lane)
  const int ty = threadIdx.y;            // 0..7  (wave in block)
  const int R  = blockIdx.x * 16;        // row tile base
  const int g  = (blockIdx.y * 8 + ty) * 16;  // col tile base
  const int n  = tx & 15;
  const int kh = tx >> 4;                // K-half select (ISA A-frag layout)

  const float* arow = X + (size_t)(R + n) * FF;  // lane row = n for A-frag
  v8f c = {};
  for (int k0 = 0; k0 < FF; k0 += 4) {
    const int k = k0 + 2 * kh;
    float2 av = *(const float2*)(arow + k);      // A[m][k], A[m][k+1]
    v2f a; a.x = av.x; a.y = av.y;
    v2f b;
    b.x = Wk[(size_t)k * FF + g + n];            // B[k][n]
    b.y = Wk[(size_t)(k + 1) * FF + g + n];      // B[k+1][n]
    c = wmma4(a, b, c);
  }
  const float bias = bk[g + n];
#pragma unroll
  for (int r = 0; r < 8; ++r) {
    const int row = r + 8 * kh;                  // C/D layout: M = r + 8*(lane>=16)
    feat[(size_t)(R + row) * FF + g + n] = c[r] + bias;
  }
}

// ---------------------------------------------------------------
// Kernel 2: a_self = feat.ws + bs ; a_neigh = feat.wn + bn  (per row)
// one wave32 per row; block = 256 threads = 8 rows; grid BN/8
// ---------------------------------------------------------------
__global__ __launch_bounds__(256) void scores_kernel(
    const float* __restrict__ feat,
    const float* __restrict__ ws, const float* __restrict__ bs,
    const float* __restrict__ wn, const float* __restrict__ bn,
    float* __restrict__ a_self, float* __restrict__ a_neigh) {
  const int lane = threadIdx.x & 31;
  const int wv   = threadIdx.x >> 5;
  const int row  = blockIdx.x * 8 + wv;
  const float4* fr  = (const float4*)(feat + (size_t)row * FF);
  const float4* wsv = (const float4*)ws;
  const float4* wnv = (const float4*)wn;
  float ss = 0.f, sn = 0.f;
#pragma unroll
  for (int cidx = 0; cidx < 4; ++cidx) {
    const int j4 = lane + 32 * cidx;             // 128 float4 = 512 floats
    const float4 f = fr[j4];
    const float4 a = wsv[j4];
    const float4 b = wnv[j4];
    ss += f.x * a.x + f.y * a.y + f.z * a.z + f.w * a.w;
    sn += f.x * b.x + f.y * b.y + f.z * b.z + f.w * b.w;
  }
  for (int off = 16; off; off >>= 1) {           // wave32 tree reduce
    ss += __shfl_xor(ss, off, 32);
    sn += __shfl_xor(sn, off, 32);
  }
  if (lane == 0) {
    a_self[row]  = ss + bs[0];
    a_neigh[row] = sn + bn[0];
  }
}

// ---------------------------------------------------------------
// Kernel 3: per-row softmax stats over masked leaky-relu scores
// att[i][j] = lrelu(as_i * an_j) + MASK*(1-A[i][j]); rmax/rsum per row
// one wave per row; grid BN/8, block 256
// ---------------------------------------------------------------
__global__ __launch_bounds__(256) void rowstats_kernel(
    const float* __restrict__ A,
    const float* __restrict__ a_self, const float* __restrict__ a_neigh,
    float* __restrict__ rmax, float* __restrict__ rsum) {
  const int lane = threadIdx.x & 31;
  const int wv   = threadIdx.x >> 5;
  const int row  = blockIdx.x * 8 + wv;          // 0..BN-1
  const int b    = row >> 11;                    // row / NN
  const float as = a_self[row];
  const float* Ar = A + (size_t)row * NN;
  const float* an = a_neigh + (size_t)b * NN;
  float mm = -3.0e38f, ssum = 0.f;
  for (int j = lane; j < NN; j += 32) {
    __builtin_prefetch(Ar + j + 256, 0, 0);      // global_prefetch on A stream
    float x = as * an[j];
    x = x > 0.f ? x : NEG_SLOPE * x;
    x += MASK_VAL * (1.0f - Ar[j]);
    const float mn = fmaxf(mm, x);
    ssum = ssum * __expf(mm - mn) + __expf(x - mn);
    mm = mn;
  }
  for (int off = 16; off; off >>= 1) {           // merge (m,s) pairs
    const float mo = __shfl_xor(mm, off, 32);
    const float so = __shfl_xor(ssum, off, 32);
    const float mn = fmaxf(mm, mo);
    ssum = ssum * __expf(mm - mn) + so * __expf(mo - mn);
    mm = mn;
  }
  if (lane == 0) { rmax[row] = mm; rsum[row] = ssum; }
}

// ---------------------------------------------------------------
// Kernel 4: out = tanh( softmax(att) @ feat ), replicated x HEADS
// block (32,8): one 16-row tile, 8 waves x 4 col-tiles = full 512 cols.
// Per K-block: TDM DMAs the feat [KBLK x FF] tile into dynamic LDS (wave 0,
// TENSORcnt) while all 256 threads build the p-hat(16 x KBLK) tile; then
// waves run f32 WMMA with A-frags from pL and B-frags from LDS feat.
// grid BN/16; dynamic LDS = KBLK*FF*4 = 64 KB
// ---------------------------------------------------------------
__global__ __launch_bounds__(256) void attn_kernel(
    const float* __restrict__ A, const float* __restrict__ feat,
    const float* __restrict__ a_self, const float* __restrict__ a_neigh,
    const float* __restrict__ rmax, const float* __restrict__ rsum,
    float* __restrict__ out) {
  extern __shared__ float ldsfeat[];             // [KBLK][FF] staged feat tile
  __shared__ float pL[16 * 33];                  // padded p-hat tile
  __shared__ float smax[16], sinv[16], sas[16];
  const int tx  = threadIdx.x;
  const int ty  = threadIdx.y;
  const int tid = ty * 32 + tx;
  const int R   = blockIdx.x * 16;               // global row base
  const int b   = R >> 11;
  if (tid < 16) {
    smax[tid] = rmax[R + tid];
    sinv[tid] = 1.0f / rsum[R + tid];
    sas[tid]  = a_self[R + tid];
  }
  __syncthreads();

  const int n  = tx & 15;
  const int kh = tx >> 4;
  v8f acc[4];
#pragma unroll
  for (int q = 0; q < 4; ++q) acc[q] = (v8f){};

  const float* anb   = a_neigh + (size_t)b * NN;
  const float* featb = feat + (size_t)b * NN * FF;
#if HAVE_TDM
  const unsigned lds_off = (unsigned)(uintptr_t)(void*)ldsfeat;  // LDS byte offset
#endif

  for (int j0 = 0; j0 < NN; j0 += KBLK) {
#if HAVE_TDM
    if (ty == 0)                                  // one TDM issue per block
      tdm_load_feat_tile(featb + (size_t)j0 * FF, lds_off);
#else
    {                                             // cooperative fallback copy
      const float4* src = (const float4*)(featb + (size_t)j0 * FF);
      float4* dst = (float4*)ldsfeat;
      for (int e = tid; e < KBLK * FF / 4; e += 256) dst[e] = src[e];
    }
#endif
    // cooperative p-hat tile: 16 rows x KBLK cols (overlaps the DMA)
#pragma unroll
    for (int e = tid; e < 16 * KBLK; e += 256) {
      const int r  = e >> 5;
      const int cc = e & 31;
      const int j  = j0 + cc;
      float x = sas[r] * anb[j];
      x = x > 0.f ? x : NEG_SLOPE * x;
      x += MASK_VAL * (1.0f - A[(size_t)(R + r) * NN + j]);
      pL[r * 33 + cc] = __expf(x - smax[r]);
    }
#if HAVE_TDM
    if (ty == 0) __builtin_amdgcn_s_wait_tensorcnt(0);
#endif
    __syncthreads();                              // publish LDS tiles

#pragma unroll
    for (int kk = 0; kk < KBLK; kk += 4) {
      v2f a;                                      // A-frag: row = n, K pair by kh
      a.x = pL[n * 33 + kk + 2 * kh];
      a.y = pL[n * 33 + kk + 2 * kh + 1];
#pragma unroll
      for (int q = 0; q < 4; ++q) {
        const int g = (ty * 4 + q) * 16;
        const float* bp = ldsfeat + (kk + 2 * kh) * FF + g + n;
        v2f bb; bb.x = bp[0]; bb.y = bp[FF];      // B[k][n], B[k+1][n] from LDS
        acc[q] = wmma4(a, bb, acc[q]);
      }
    }
    __syncthreads();                              // tiles consumed; safe to refill
  }

  // epilogue: scale by 1/rowsum, tanh, replicate across HEADS
#pragma unroll
  for (int q = 0; q < 4; ++q) {
    const int g = (ty * 4 + q) * 16;
#pragma unroll
    for (int r = 0; r < 8; ++r) {
      const int row = r + 8 * kh;
      const float t = tanhf(acc[q][r] * sinv[row]);
      float* op = out + (size_t)(R + row) * (HEADS * FF) + g + n;
#pragma unroll
      for (int h = 0; h < HEADS; ++h) op[h * FF] = t;
    }
  }
}

// ---------------------------------------------------------------
extern "C" void kernel_launch(void* const* d_in, const int* in_sizes, int n_in,
                              void* d_out, int out_size, void* d_ws, size_t ws_size,
                              hipStream_t stream) {
  const float* X  = (const float*)d_in[0];   // [B,N,F]
  const float* A  = (const float*)d_in[1];   // [B,N,N]
  const float* Wk = (const float*)d_in[2];   // [F,F]
  const float* bk = (const float*)d_in[3];   // [F]
  const float* ws = (const float*)d_in[4];   // [F]
  const float* bs = (const float*)d_in[5];   // [1]
  const float* wn = (const float*)d_in[6];   // [F]
  const float* bn = (const float*)d_in[7];   // [1]
  float* out = (float*)d_out;                // [B,N,HEADS*F]

  char* w = (char*)d_ws;
  float* feat    = (float*)w; w += (size_t)BN * FF * sizeof(float);  // 32 MB
  float* a_self  = (float*)w; w += (size_t)BN * sizeof(float);
  float* a_neigh = (float*)w; w += (size_t)BN * sizeof(float);
  float* rmax    = (float*)w; w += (size_t)BN * sizeof(float);
  float* rsum    = (float*)w; w += (size_t)BN * sizeof(float);

  proj_kernel<<<dim3(BN / 16, FF / 128), dim3(32, 8), 0, stream>>>(X, Wk, bk, feat);
  scores_kernel<<<dim3(BN / 8), 256, 0, stream>>>(feat, ws, bs, wn, bn, a_self, a_neigh);
  rowstats_kernel<<<dim3(BN / 8), 256, 0, stream>>>(A, a_self, a_neigh, rmax, rsum);
  attn_kernel<<<dim3(BN / 16), dim3(32, 8), (size_t)(KBLK * FF * sizeof(float)),
               stream>>>(A, feat, a_self, a_neigh, rmax, rsum, out);
}